// HierarchicalMultinomialRegression_62285615727250
// MI455X (gfx1250) — compile-verified
//
#include <hip/hip_runtime.h>
#include <math.h>

typedef float v2f __attribute__((ext_vector_type(2)));
typedef float v8f __attribute__((ext_vector_type(8)));

#define NF   64   // fixed-effect columns of X
#define KOUT 8    // output columns (beta cols, K)
#define BDIM 5000
#define LDIM 10
#define TDIM 25

// ---------------------------------------------------------------------------
// Kernel 1: rho = tanh(raw_rho)  (L,K);  d = diag(cholesky(Sigma)).T  (L,K)
// Sigma is (K, L, L). One thread per k does a tiny serial Cholesky.
// ---------------------------------------------------------------------------
__global__ void prep_kernel(const float* __restrict__ raw_rho,
                            const float* __restrict__ Sigma,
                            float* __restrict__ rho_out,
                            float* __restrict__ d_out)
{
    int tid = threadIdx.x;
    if (tid < LDIM * KOUT) rho_out[tid] = tanhf(raw_rho[tid]);
    if (tid >= 80 && tid < 80 + KOUT) {
        int k = tid - 80;
        const float* S = Sigma + k * LDIM * LDIM;
        float Lm[LDIM][LDIM];
        for (int j = 0; j < LDIM; j++) {
            float s = S[j * LDIM + j];
            for (int p = 0; p < j; p++) s -= Lm[j][p] * Lm[j][p];
            float dj = sqrtf(s);
            Lm[j][j] = dj;
            float inv = 1.0f / dj;
            for (int i = j + 1; i < LDIM; i++) {
                float t = S[i * LDIM + j];
                for (int p = 0; p < j; p++) t -= Lm[i][p] * Lm[j][p];
                Lm[i][j] = t * inv;
            }
            d_out[j * KOUT + k] = dj;   // d[l][k] = SigmaHalf[k][l][l]
        }
    }
}

// ---------------------------------------------------------------------------
// Kernel 2: AR(1)-style scan over t for each (b,l) pair.
//   u[b,l,0,:]  = eps[b,l,0,:]
//   u[b,l,t,:]  = rho[l,:] * u[b,l,t-1,:] + d[l,:] * eps[b,l,t,:]
// eps layout (B,L,T,K) is contiguous per (b,l): 200 floats per thread.
// u stored in ws with the same (b,l,t,k) layout for the later gather.
// ---------------------------------------------------------------------------
__global__ void scan_kernel(const float* __restrict__ eps,
                            const float* __restrict__ rho,
                            const float* __restrict__ dvec,
                            float* __restrict__ u)
{
    int i = blockIdx.x * blockDim.x + threadIdx.x;
    if (i >= BDIM * LDIM) return;
    int l = i % LDIM;
    float rr[KOUT], dd[KOUT], uc[KOUT];
#pragma unroll
    for (int k = 0; k < KOUT; k++) { rr[k] = rho[l * KOUT + k]; dd[k] = dvec[l * KOUT + k]; }
    const float* e  = eps + (size_t)i * (TDIM * KOUT);
    float*       uo = u   + (size_t)i * (TDIM * KOUT);
#pragma unroll
    for (int k = 0; k < KOUT; k++) { uc[k] = e[k]; uo[k] = uc[k]; }
    for (int t = 1; t < TDIM; t++) {
#pragma unroll
        for (int k = 0; k < KOUT; k++) {
            uc[k] = rr[k] * uc[k] + dd[k] * e[t * KOUT + k];
            uo[t * KOUT + k] = uc[k];
        }
    }
}

// ---------------------------------------------------------------------------
// Kernel 3: out = X @ beta + u[gather]
// One wave (32 lanes) per 16-row tile of X. 8 waves / 256-thread block.
// A fragments via LDS (row stride 68 floats -> conflict-free v2f reads),
// B fragments (beta zero-padded 64x16) in registers, C accumulated with
// 16x V_WMMA_F32_16X16X4_F32, then remapped through LDS for coalesced store.
// ---------------------------------------------------------------------------
#define XP 68           // padded LDS row stride for the X tile
#define WAVES 8

__global__ __launch_bounds__(256)
void gemm_gather_kernel(const float* __restrict__ X,
                        const float* __restrict__ beta,
                        const int*  __restrict__ bat,
                        const int*  __restrict__ lea,
                        const int*  __restrict__ sea,
                        const float* __restrict__ u,
                        float* __restrict__ out,
                        int ntiles)
{
    __shared__ __align__(16) float xs[WAVES][16 * XP];
    __shared__ __align__(16) float cs[WAVES][16 * 17];
    __shared__ int is[WAVES][16];

    const int lane  = threadIdx.x & 31;
    const int wave  = threadIdx.x >> 5;
    const int tile  = blockIdx.x * WAVES + wave;
    const bool active = tile < ntiles;           // wave-uniform
    const int rowbase = tile * 16;

    const int n     = lane & 15;                 // WMMA column / row-in-tile id
    const int khalf = lane >> 4;                 // 0: K{0,1}, 1: K{2,3} per step

    // ---- B fragments: beta (64 x 8) zero-padded to 64 x 16, per ISA layout:
    // b[j] = B[4*s + 2*khalf + j][n]
    v2f bfrag[16];
#pragma unroll
    for (int s = 0; s < 16; s++) {
        int kk = 4 * s + 2 * khalf;
        v2f b;
        b.x = (n < KOUT) ? beta[kk * KOUT + n]       : 0.0f;
        b.y = (n < KOUT) ? beta[(kk + 1) * KOUT + n] : 0.0f;
        bfrag[s] = b;
    }

    // ---- Stage X tile into LDS (coalesced float4) + per-row gather index
    if (active) {
        const float* Xt = X + (size_t)rowbase * NF;
#pragma unroll
        for (int it = 0; it < 8; it++) {
            int idx = it * 128 + lane * 4;       // 0..1023, 16B per lane
            int r = idx >> 6, c = idx & 63;
            float4 v = *(const float4*)(Xt + idx);
            float* dst = &xs[wave][r * XP + c];
            dst[0] = v.x; dst[1] = v.y; dst[2] = v.z; dst[3] = v.w;
        }
        if (lane < 16) {
            int m = rowbase + lane;
            // u index base: ((b*L + l)*T + t)*K, matches ws (b,l,t,k) layout
            is[wave][lane] = ((bat[m] * LDIM + lea[m]) * TDIM + sea[m]) * KOUT;
        }
        // prefetch next tile this wave family will touch
        if (tile + WAVES < ntiles)
            __builtin_prefetch(X + (size_t)(rowbase + 16 * WAVES) * NF + lane * 16, 0, 1);
    }
    __syncthreads();

    // ---- 16 x V_WMMA_F32_16X16X4_F32 over the K=64 reduction
    v8f c = {};
    if (active) {
        const float* xrow = &xs[wave][n * XP + 2 * khalf];  // A: m=n(lane%16)
#pragma unroll
        for (int s = 0; s < 16; s++) {
            v2f a = *(const v2f*)(xrow + 4 * s);
            c = __builtin_amdgcn_wmma_f32_16x16x4_f32(
                    /*neg_a=*/false, a, /*neg_b=*/false, bfrag[s],
                    /*c_mod=*/(short)0, c, /*reuse_a=*/false, /*reuse_b=*/false);
        }
        // C layout: vgpr r holds row m = r + 8*khalf, col = n
#pragma unroll
        for (int r = 0; r < 8; r++)
            cs[wave][(r + 8 * khalf) * 17 + n] = c[r];
    }
    __syncthreads();

    // ---- Coalesced epilogue: out[row,col] = C + u[gather]
    if (active) {
        float* outt = out + (size_t)rowbase * KOUT;
#pragma unroll
        for (int q = 0; q < 4; q++) {
            int j   = lane + 32 * q;       // 0..127 over 16 rows x 8 cols
            int r   = j >> 3;
            int col = j & 7;
            float uval = u[is[wave][r] + col];
            outt[j] = cs[wave][r * 17 + col] + uval;
        }
    }
}

// ---------------------------------------------------------------------------
// Host launcher. Input order (setup_inputs dict):
//   0:X(f32 N x64) 1:batter_ids(i32) 2:league_ids(i32) 3:season_ids(i32)
//   4:beta(f32 64x8) 5:raw_rho(f32 10x8) 6:Sigma(f32 8x10x10) 7:eps(f32 B,L,T,K)
// Output: (N, 8) f32.
// Workspace: [0..127] rho, [128..255] d, [256 ...] u (10M floats = 40 MB).
// ---------------------------------------------------------------------------
extern "C" void kernel_launch(void* const* d_in, const int* in_sizes, int n_in,
                              void* d_out, int out_size, void* d_ws, size_t ws_size,
                              hipStream_t stream)
{
    const float* X       = (const float*)d_in[0];
    const int*   bat     = (const int*)  d_in[1];
    const int*   lea     = (const int*)  d_in[2];
    const int*   sea     = (const int*)  d_in[3];
    const float* beta    = (const float*)d_in[4];
    const float* raw_rho = (const float*)d_in[5];
    const float* Sigma   = (const float*)d_in[6];
    const float* eps     = (const float*)d_in[7];
    float* out = (float*)d_out;

    float* rho_w = (float*)d_ws;
    float* dv_w  = rho_w + 128;
    float* u_w   = rho_w + 256;

    const int N = in_sizes[0] / NF;

    prep_kernel<<<1, 96, 0, stream>>>(raw_rho, Sigma, rho_w, dv_w);

    int nscan = BDIM * LDIM;
    scan_kernel<<<(nscan + 255) / 256, 256, 0, stream>>>(eps, rho_w, dv_w, u_w);

    int ntiles = N / 16;                        // N = 1,000,000 -> 62,500
    int blocks = (ntiles + WAVES - 1) / WAVES;
    gemm_gather_kernel<<<blocks, 256, 0, stream>>>(X, beta, bat, lea, sea, u_w, out, ntiles);
}